// PatchMatchAttentionMARINER_42391327211609
// MI455X (gfx1250) — compile-verified
//
#include <hip/hip_runtime.h>
#include <hip/hip_bf16.h>
#include <math.h>

typedef float v2f __attribute__((ext_vector_type(2)));
typedef float v8f __attribute__((ext_vector_type(8)));

#define NN 2
#define C1 128
#define HH 64
#define WW 64
#define NBLK 64          // blocks per image (8x8)
#define NB_ (NN*NBLK)    // 128
#define FF 1152          // C1*3*3
#define MM 4096          // 64*64 coarse candidates
#define FINE_M 169       // 13*13 fine candidates

// ---------------- coarse: build normalized LR center-patch features ----------------
__global__ void k_build_lr(const float* __restrict__ x1, float* __restrict__ lr)
{
    int t = blockIdx.x*128 + threadIdx.x;
    if (t >= NN*3*NBLK) return;
    int n   = t / (3*NBLK);
    int rem = t % (3*NBLK);
    int d   = rem / NBLK;
    int b   = rem % NBLK;
    int dil = d + 1;
    int by = b >> 3, bx = b & 7;
    int r0 = by*8 + 4, c0 = bx*8 + 4;    // dilated 3x3 center, never hits padding
    const float* base = x1 + (size_t)n*C1*HH*WW;
    float ss = 0.f;
    for (int c = 0; c < C1; ++c)
      for (int dy = -1; dy <= 1; ++dy)
        for (int dx = -1; dx <= 1; ++dx) {
            float v = base[(c*HH + (r0 + dy*dil))*WW + (c0 + dx*dil)];
            ss += v*v;
        }
    float inv = 1.f / fmaxf(sqrtf(ss), 1e-12f);
    float* out = lr + (size_t)((n*3 + d)*NBLK + b)*FF;
    for (int c = 0; c < C1; ++c)
      for (int dy = -1; dy <= 1; ++dy)
        for (int dx = -1; dx <= 1; ++dx) {
            float v = base[(c*HH + (r0 + dy*dil))*WW + (c0 + dx*dil)];
            out[c*9 + (dy+1)*3 + (dx+1)] = v*inv;
        }
}

// ---------------- coarse: per-column inverse L2 norm of ref patches ----------------
__global__ void k_colnorm(const float* __restrict__ ref1, int n, int dil,
                          float* __restrict__ invn)
{
    int m = blockIdx.x*256 + threadIdx.x;
    if (m >= MM) return;
    int y = m >> 6, x = m & 63;
    const float* base = ref1 + (size_t)n*C1*HH*WW;
    float ss = 0.f;
    for (int c = 0; c < C1; ++c)
      for (int ky = 0; ky < 3; ++ky) {
          int ry = y + (ky-1)*dil;
          if (ry < 0 || ry >= HH) continue;
          for (int kx = 0; kx < 3; ++kx) {
              int rx = x + (kx-1)*dil;
              if (rx < 0 || rx >= WW) continue;
              float v = base[(c*HH + ry)*WW + rx];
              ss += v*v;
          }
      }
    invn[m] = 1.f / fmaxf(sqrtf(ss), 1e-12f);
}

// ---------------- coarse: materialize normalized patch matrix rp[f][m] ----------------
__global__ void k_rpfill(const float* __restrict__ ref1, int n, int dil,
                         const float* __restrict__ invn, float* __restrict__ rp)
{
    int g = blockIdx.x*256 + threadIdx.x;
    if (g >= FF*MM) return;
    int f = g >> 12;          // /4096
    int m = g & 4095;
    int c = f / 9, r9 = f % 9, ky = r9/3, kx = r9%3;
    int y = m >> 6, x = m & 63;
    int ry = y + (ky-1)*dil, rx = x + (kx-1)*dil;
    float v = 0.f;
    if (ry >= 0 && ry < HH && rx >= 0 && rx < WW)
        v = ref1[((size_t)n*C1 + c)*HH*WW + (size_t)ry*WW + rx];
    rp[g] = v * invn[m];
}

// ---------------- coarse GEMM: (64 x 1152) * (1152 x 4096), fp32 WMMA, accumulate ----
__global__ void k_cgemm(const float* __restrict__ A, const float* __restrict__ Bm,
                        float* __restrict__ Cm, int first)
{
    int wv   = blockIdx.x*8 + (threadIdx.x >> 5);
    int rt   = wv >> 8;           // 4 row tiles
    int ct   = wv & 255;          // 256 col tiles
    int lane = threadIdx.x & 31;
    int half = lane >> 4;         // 0: K 0,1   1: K 2,3
    int l15  = lane & 15;
    int sel  = half << 1;
    int row  = rt*16 + l15;
    int col  = ct*16 + l15;
    v8f acc;
    if (first) {
        for (int i = 0; i < 8; ++i) acc[i] = 0.f;
    } else {
        for (int i = 0; i < 8; ++i)
            acc[i] = Cm[(size_t)(rt*16 + i + half*8)*MM + col];
    }
    const float* arow = A + (size_t)row*FF;
    for (int k = 0; k < FF; k += 4) {
        v2f a, b;
        a.x = arow[k + sel];
        a.y = arow[k + sel + 1];
        b.x = Bm[(size_t)(k + sel    )*MM + col];
        b.y = Bm[(size_t)(k + sel + 1)*MM + col];
        acc = __builtin_amdgcn_wmma_f32_16x16x4_f32(false, a, false, b,
                                                    (short)0, acc, false, false);
    }
    for (int i = 0; i < 8; ++i)
        Cm[(size_t)(rt*16 + i + half*8)*MM + col] = acc[i];
}

// ---------------- coarse argmax + clamped search-window computation ----------------
__global__ void k_argmax_window(const float* __restrict__ corr,
                                int* __restrict__ idxh1, int* __restrict__ idxw1)
{
    int t = threadIdx.x;
    if (t >= NB_) return;
    const float* row = corr + (size_t)t*MM;
    float best = row[0]; int bi = 0;
    for (int m = 1; m < MM; ++m) { float v = row[m]; if (v > best) { best = v; bi = m; } }
    int iw = bi & 63, ih = bi >> 6;
    int w1 = iw - 7, w2 = iw + 7;
    if (w1 < 0)  { w1 = 0;  w2 = 14; }
    if (w2 > 63) { w2 = 63; w1 = 49; }
    int h1 = ih - 7, h2 = ih + 7;
    if (h1 < 0)  { h1 = 0;  h2 = 14; }
    if (h2 > 63) { h2 = 63; h1 = 49; }
    idxw1[t] = w1; idxh1[t] = h1;
}

// ---------------- fine search: one workgroup per block, LDS-resident --------------
__global__ void k_fine(const float* __restrict__ x1, const float* __restrict__ ref1,
                       const int* __restrict__ idxh1, const int* __restrict__ idxw1,
                       float* __restrict__ soft, int* __restrict__ fidx)
{
    extern __shared__ float sm[];
    float* rblk = sm;                  // 128*10*10 = 12800
    float* refb = rblk + C1*100;       // 128*15*15 = 28800
    float* invA = refb + C1*225;       // 64
    float* invB = invA + 64;           // 176 (padded)
    float* tbuf = invB + 176;          // 4 waves * 256

    int nb = blockIdx.x;
    int n = nb >> 6, b = nb & 63;
    int by = b >> 3, bx = b & 7;
    int tid = threadIdx.x;             // 128 threads, 4 waves

    const float* xb = x1 + (size_t)n*C1*HH*WW;
    for (int i = tid; i < C1*100; i += 128) {
        int c = i / 100, rr = (i % 100)/10, cc = i % 10;
        int ry = by*8 - 1 + rr; ry = min(max(ry, 0), HH-1);  // 'edge' pad
        int rx = bx*8 - 1 + cc; rx = min(max(rx, 0), WW-1);
        rblk[i] = xb[(c*HH + ry)*WW + rx];
    }
    int h1 = idxh1[nb], w1 = idxw1[nb];
    const float* rb = ref1 + (size_t)n*C1*HH*WW;
    for (int i = tid; i < C1*225; i += 128) {
        int c = i / 225, rr = (i % 225)/15, cc = i % 15;
        refb[i] = rb[(c*HH + (h1 + rr))*WW + (w1 + cc)];
    }
    __syncthreads();

    if (tid < 64) {
        int oy = tid >> 3, ox = tid & 7;
        float ss = 0.f;
        for (int c = 0; c < C1; ++c)
          for (int ky = 0; ky < 3; ++ky)
            for (int kx = 0; kx < 3; ++kx) {
                float v = rblk[c*100 + (oy+ky)*10 + (ox+kx)];
                ss += v*v;
            }
        invA[tid] = 1.f / fmaxf(sqrtf(ss), 1e-12f);
    }
    for (int m = tid; m < FINE_M; m += 128) {
        int py = m / 13, px = m % 13;
        float ss = 0.f;
        for (int c = 0; c < C1; ++c)
          for (int ky = 0; ky < 3; ++ky)
            for (int kx = 0; kx < 3; ++kx) {
                float v = refb[c*225 + (py+ky)*15 + (px+kx)];
                ss += v*v;
            }
        invB[m] = 1.f / fmaxf(sqrtf(ss), 1e-12f);
    }
    __syncthreads();

    int wv   = tid >> 5;              // row tile 0..3
    int lane = tid & 31;
    int half = lane >> 4;
    int l15  = lane & 15;
    int sel  = half << 1;
    int rowA = wv*16 + l15;
    int oy = rowA >> 3, ox = rowA & 7;
    float best = -3.4e38f; int bi = 0;

    for (int ct = 0; ct < 11; ++ct) {
        int col   = ct*16 + l15;
        int colOK = (col < FINE_M);
        int py = colOK ? col/13 : 0;
        int px = colOK ? col%13 : 0;
        v8f acc;
        for (int i = 0; i < 8; ++i) acc[i] = 0.f;
        for (int k = 0; k < FF; k += 4) {
            int k0 = k + sel, k1 = k0 + 1;
            int c0 = k0/9, t0 = k0%9, ky0 = t0/3, kx0 = t0%3;
            int c1 = k1/9, t1 = k1%9, ky1 = t1/3, kx1 = t1%3;
            v2f a, bb;
            a.x  = rblk[c0*100 + (oy+ky0)*10 + (ox+kx0)];
            a.y  = rblk[c1*100 + (oy+ky1)*10 + (ox+kx1)];
            bb.x = colOK ? refb[c0*225 + (py+ky0)*15 + (px+kx0)] : 0.f;
            bb.y = colOK ? refb[c1*225 + (py+ky1)*15 + (px+kx1)] : 0.f;
            acc = __builtin_amdgcn_wmma_f32_16x16x4_f32(false, a, false, bb,
                                                        (short)0, acc, false, false);
        }
        for (int i = 0; i < 8; ++i)
            tbuf[wv*256 + (i + half*8)*16 + l15] = acc[i];
        __syncthreads();   // uniform; orders LDS tile for the cross-lane scan
        if (lane < 16) {
            for (int cl = 0; cl < 16; ++cl) {
                int cc = ct*16 + cl;
                if (cc >= FINE_M) break;
                float v = tbuf[wv*256 + lane*16 + cl] * invB[cc];
                if (v > best) { best = v; bi = cc; }   // first max, like jnp.argmax
            }
        }
        __syncthreads();
    }
    if (lane < 16) {
        int rg = wv*16 + lane;
        soft[(size_t)nb*64 + rg] = best * invA[rg];
        fidx[(size_t)nb*64 + rg] = bi;
    }
}

// ---------------- transfer: gather + overlap-add (analytic divisor) + bilinear att --
__global__ void k_transfer(const float* __restrict__ ref, float* __restrict__ outp,
                           const float* __restrict__ soft, const int* __restrict__ fidx,
                           const int* __restrict__ idxh1, const int* __restrict__ idxw1,
                           int C, int s, int total)
{
    int g = blockIdx.x*256 + threadIdx.x;
    if (g >= total) return;
    int HW = 64*s, bs = 8*s, ks = 3*s;
    int x = g % HW; int t1 = g / HW;
    int y = t1 % HW; int t2 = t1 / HW;
    int c = t2 % C;  int n = t2 / C;
    int by = y / bs, ho = y % bs;
    int bx = x / bs, wo = x % bs;
    int nb = n*64 + by*8 + bx;
    int ih1 = idxh1[nb], iw1 = idxw1[nb];
    int hp = ho + s, wp = wo + s;         // pd = s

    int his[3], kys[3], wis[3], kxs[3];
    int cnth = 0, cntw = 0;
    for (int hh = 0; hh < 8; ++hh) {
        int ky = hp - hh*s;
        if (ky >= 0 && ky < ks) { his[cnth] = hh; kys[cnth] = ky; ++cnth; }
        int kx = wp - hh*s;
        if (kx >= 0 && kx < ks) { wis[cntw] = hh; kxs[cntw] = kx; ++cntw; }
    }
    const float* rbase = ref + ((size_t)n*C + c)*HW*HW;
    const int* fi = fidx + (size_t)nb*64;
    float sum = 0.f;
    for (int i = 0; i < cnth; ++i)
      for (int j = 0; j < cntw; ++j) {
          int m = fi[his[i]*8 + wis[j]];
          int py = m/13, px = m%13;
          int rr = (ih1 + py)*s + kys[i];
          int cc = (iw1 + px)*s + kxs[j];
          sum += rbase[(size_t)rr*HW + cc];
      }
    float val = sum / (float)(cnth*cntw);

    const float* S = soft + (size_t)nb*64;
    float fy = (ho + 0.5f)/(float)s - 0.5f;
    float fx = (wo + 0.5f)/(float)s - 0.5f;
    int y0 = (int)floorf(fy); float wy = fy - (float)y0;
    int x0 = (int)floorf(fx); float wx = fx - (float)x0;
    int ya = min(max(y0, 0), 7), yb2 = min(max(y0+1, 0), 7);
    int xa = min(max(x0, 0), 7), xb2 = min(max(x0+1, 0), 7);
    float att = (1.f-wy)*((1.f-wx)*S[ya*8+xa]  + wx*S[ya*8+xb2])
              +      wy *((1.f-wx)*S[yb2*8+xa] + wx*S[yb2*8+xb2]);

    outp[((size_t)n*C + c)*HW*HW + (size_t)y*HW + x] = val * att;
}

extern "C" void kernel_launch(void* const* d_in, const int* in_sizes, int n_in,
                              void* d_out, int out_size, void* d_ws, size_t ws_size,
                              hipStream_t stream)
{
    (void)in_sizes; (void)n_in; (void)out_size; (void)ws_size;
    const float* r1   = (const float*)d_in[2];   // fea_r_x1   (2,128,64,64)
    const float* ref4 = (const float*)d_in[3];   // fea_ref_x4 (2,32,256,256)
    const float* ref2 = (const float*)d_in[4];   // fea_ref_x2 (2,64,128,128)
    const float* ref1 = (const float*)d_in[5];   // fea_ref_x1 (2,128,64,64)

    float* out  = (float*)d_out;
    float* out4 = out;
    float* out2 = out4 + (size_t)2*32*256*256;
    float* out1 = out2 + (size_t)2*64*128*128;

    float* ws = (float*)d_ws;
    size_t o = 0;
    float* lr   = ws + o;  o += (size_t)NN*3*NBLK*FF;   // 442368
    float* rp   = ws + o;  o += (size_t)FF*MM;          // 4718592
    float* invn = ws + o;  o += MM;                     // 4096
    float* corr = ws + o;  o += (size_t)NB_*MM;         // 524288
    int* idxh1  = (int*)(ws + o);  o += 128;
    int* idxw1  = (int*)(ws + o);  o += 128;
    float* soft = ws + o;  o += (size_t)NB_*64;         // 8192
    int* fidx   = (int*)(ws + o);                       // 8192 ints

    k_build_lr<<<3, 128, 0, stream>>>(r1, lr);

    for (int n = 0; n < NN; ++n) {
        for (int di = 0; di < 3; ++di) {
            int dil = di + 1;
            k_colnorm<<<16, 256, 0, stream>>>(ref1, n, dil, invn);
            k_rpfill<<<(FF*MM + 255)/256, 256, 0, stream>>>(ref1, n, dil, invn, rp);
            k_cgemm<<<128, 256, 0, stream>>>(lr + (size_t)(n*3 + di)*NBLK*FF, rp,
                                             corr + (size_t)n*NBLK*MM, di == 0);
        }
    }

    k_argmax_window<<<1, 128, 0, stream>>>(corr, idxh1, idxw1);

    size_t fsm = (size_t)(C1*100 + C1*225 + 64 + 176 + 4*256) * sizeof(float); // ~172 KB LDS
    k_fine<<<NB_, 128, fsm, stream>>>(r1, ref1, idxh1, idxw1, soft, fidx);

    int tot4 = 2*32*256*256;
    int tot2 = 2*64*128*128;
    int tot1 = 2*128*64*64;
    k_transfer<<<(tot4 + 255)/256, 256, 0, stream>>>(ref4, out4, soft, fidx, idxh1, idxw1, 32, 4, tot4);
    k_transfer<<<(tot2 + 255)/256, 256, 0, stream>>>(ref2, out2, soft, fidx, idxh1, idxw1, 64, 2, tot2);
    k_transfer<<<(tot1 + 255)/256, 256, 0, stream>>>(ref1, out1, soft, fidx, idxh1, idxw1, 128, 1, tot1);
}